// LIF_88794153877970
// MI455X (gfx1250) — compile-verified
//
#include <hip/hip_runtime.h>

// LIF forward (ZIF spike), fp32, [B, T, N] with T-sequential recurrence.
// Memory-bound streaming kernel: 128-bit non-temporal loads/stores, all T=16
// loads issued up front per thread for maximum memory-level parallelism.
// N is a compile-time power of two -> index split is shift/mask (no int div).

typedef float v4f __attribute__((ext_vector_type(4)));

#define LIF_T       16
#define LIF_N       65536
#define LIF_TAU     0.25f
#define LIF_NVEC    (LIF_N / 4)        // 16384 = 1 << 14
#define LIF_NVEC_SH 14
#define LIF_NVEC_MS (LIF_NVEC - 1)

__global__ __launch_bounds__(256) void lif_fwd_kernel(
    const v4f* __restrict__ xv,      // x as float4 vectors
    const float* __restrict__ thresh,
    v4f* __restrict__ ov,            // spikes as float4 vectors
    int total_vec)                   // B * N/4
{
    int i = blockIdx.x * blockDim.x + threadIdx.x;
    if (i >= total_vec) return;

    int b = i >> LIF_NVEC_SH;                  // batch index
    int v = i & LIF_NVEC_MS;                   // vector index within N
    int base = b * (LIF_T * LIF_NVEC) + v;     // [b, t=0, 4*v]

    const float th = thresh[0];  // uniform scalar threshold

    // Issue all T loads first: 16 outstanding global_load_b128 per thread
    // (stride N floats = 256 KiB between time steps; contiguous across lanes).
    v4f xs[LIF_T];
#pragma unroll
    for (int t = 0; t < LIF_T; ++t) {
        xs[t] = __builtin_nontemporal_load(xv + base + t * LIF_NVEC);
    }

    // Sequential LIF recurrence (short dependence chain, 16 steps).
    v4f mem = (v4f)0.0f;
#pragma unroll
    for (int t = 0; t < LIF_T; ++t) {
        mem = mem * LIF_TAU + xs[t];     // decay + integrate (v_fmac)

        v4f sp;
        sp.x = (mem.x > th) ? 1.0f : 0.0f;
        sp.y = (mem.y > th) ? 1.0f : 0.0f;
        sp.z = (mem.z > th) ? 1.0f : 0.0f;
        sp.w = (mem.w > th) ? 1.0f : 0.0f;

        // multiplicative reset: (1 - spike) * mem  ==  spike ? 0 : mem
        mem.x = (mem.x > th) ? 0.0f : mem.x;
        mem.y = (mem.y > th) ? 0.0f : mem.y;
        mem.z = (mem.z > th) ? 0.0f : mem.z;
        mem.w = (mem.w > th) ? 0.0f : mem.w;

        __builtin_nontemporal_store(sp, ov + base + t * LIF_NVEC);
    }
}

extern "C" void kernel_launch(void* const* d_in, const int* in_sizes, int n_in,
                              void* d_out, int out_size, void* d_ws, size_t ws_size,
                              hipStream_t stream) {
    const float* x      = (const float*)d_in[0];   // [B, T, N] fp32
    const float* thresh = (const float*)d_in[1];   // [1] fp32
    float* out          = (float*)d_out;           // [B, T, N] fp32

    const long long total = (long long)in_sizes[0];        // B*T*N
    const int B = (int)(total / ((long long)LIF_T * LIF_N));
    const int total_vec = B * LIF_NVEC;                    // B*N/4

    const int block = 256;
    const int grid = (total_vec + block - 1) / block;

    lif_fwd_kernel<<<grid, block, 0, stream>>>(
        (const v4f*)x, thresh, (v4f*)out, total_vec);
}